// jacobian_map_26706106646783
// MI455X (gfx1250) — compile-verified
//
#include <hip/hip_runtime.h>

// Fused Jacobian-determinant regularizer for (1,3,192,192,192) f32 field.
// Bandwidth-bound stencil: one pass, async global->LDS slice pipeline (3-slot
// ring, ASYNCcnt), register z-ring for filtered sums. Output: 191^3 f32.

#define NV 192          // input spatial extent
#define NG 191          // gradient / output extent
#define TS 16           // tile size in y,x
#define CH 24           // z-chunk depth per workgroup
#define USTR 20         // LDS row stride for u slices (19 wide + pad)
#define GSTR 19         // LDS row stride for gradient slices (18 wide + pad)

#define AS1 __attribute__((address_space(1)))
#define AS3 __attribute__((address_space(3)))

// ---- CDNA5 async global->LDS copy (ASYNCcnt) ------------------------------
__device__ __forceinline__ void async_copy_f32(const float* gp, float* lp)
{
#if __has_builtin(__builtin_amdgcn_global_load_async_to_lds_b32)
    // ROCm clang signature uses typed pointers: (as1 int*, as3 int*, Ii, Ii)
    __builtin_amdgcn_global_load_async_to_lds_b32(
        (AS1 int*)gp, (AS3 int*)lp, /*offset=*/0, /*cpol=*/0);
#else
    AS3 float* l3 = (AS3 float*)lp;
    asm volatile("global_load_async_to_lds_b32 %0, %1, off"
                 :: "v"(l3), "v"(gp) : "memory");
#endif
}

__device__ __forceinline__ void wait_async0()
{
#if __has_builtin(__builtin_amdgcn_s_wait_asynccnt)
    __builtin_amdgcn_s_wait_asynccnt(0);
#else
    asm volatile("s_wait_asynccnt 0x0" ::: "memory");
#endif
}

// Issue async load of one 19x19(+halo) u slice (3 channels) into ring slot
// ss%3. OOB halo lanes store 0.0 directly (plain ds_store).
__device__ __forceinline__ void issue_slice_load(const float* __restrict__ u,
                                                 float* __restrict__ su,
                                                 int ss, int oy, int ox, int tid)
{
    const int slot = ss % 3;
#pragma unroll
    for (int k = 0; k < 5; ++k) {
        int i = tid + k * 256;
        if (i < 3 * 19 * 19) {
            int c  = i / (19 * 19);
            int rr = i - c * (19 * 19);
            int uy = rr / 19;
            int ux = rr - uy * 19;
            int gy = oy - 1 + uy;
            int gx = ox - 1 + ux;
            float* lp = &su[((c * 3 + slot) * 19 + uy) * USTR + ux];
            if (gy >= 0 && gy < NV && gx >= 0 && gx < NV) {
                const float* gp = &u[(((size_t)c * NV + ss) * NV + gy) * NV + gx];
                async_copy_f32(gp, lp);
            } else {
                *lp = 0.0f;
            }
        }
    }
}

// Prefetch hint (gfx1250: global_prefetch_b8) for slice s2, one row per lane.
__device__ __forceinline__ void prefetch_slice(const float* __restrict__ u,
                                               int s2, int oy, int ox, int tid)
{
    if (tid < 57) {
        int c  = tid / 19;
        int uy = tid - c * 19;
        int gy = oy - 1 + uy;
        if (gy < 0) gy = 0;
        if (gy > NV - 1) gy = NV - 1;
        int gx = ox - 1;
        if (gx < 0) gx = 0;
        const float* p = &u[(((size_t)c * NV + s2) * NV + gy) * NV + gx];
        __builtin_prefetch(p, 0, 1);
        __builtin_prefetch(p + 16, 0, 1);
    }
}

__global__ __launch_bounds__(256) void jacobian_det_kernel(
    const float* __restrict__ u, float* __restrict__ out)
{
    // u slices: 3 channels x 3 ring slots x 19 rows (stride USTR)
    __shared__ float su[3 * 3 * 19 * USTR];
    // gradient fields: 9 fields x 18 rows (stride GSTR); f = c*3 + d,
    // d: 0 = D-shift (dy), 1 = H-shift (dx), 2 = W-shift (dz)
    __shared__ float sg[9 * 18 * GSTR];

    const int tx  = threadIdx.x;
    const int ty  = threadIdx.y;
    const int tid = ty * TS + tx;
    const int ox  = blockIdx.x * TS;
    const int oy  = blockIdx.y * TS;
    const int z0  = blockIdx.z * CH;
    const int zEnd = ((z0 + CH) < NG ? (z0 + CH) : NG) - 1;   // inclusive

    const int y = oy + ty;
    const int x = ox + tx;
    const bool outValid = (y < NG) && (x < NG);

    // z-ring of 2D-filtered sums: Hm2 = H_{s-2}, Hm1 = H_{s-1}
    float Hm2[9], Hm1[9];
#pragma unroll
    for (int f = 0; f < 9; ++f) { Hm2[f] = 0.0f; Hm1[f] = 0.0f; }

    // preload: first valid gradient slice needs u slices sFirst, sFirst+1
    const int sFirst = (z0 > 0) ? (z0 - 1) : 0;
    issue_slice_load(u, su, sFirst,     oy, ox, tid);
    issue_slice_load(u, su, sFirst + 1, oy, ox, tid);

    for (int s = z0 - 1; s <= zEnd + 1; ++s) {
        float Hc[9];
        const bool sValid = (s >= 0) && (s < NG);
        if (sValid) {
            // slices s and s+1 were issued earlier: wait for them to land.
            wait_async0();
            __syncthreads();

            // pipeline: issue slice s+2 now; it transfers while we compute.
            if (s <= zEnd && s + 1 < NG)
                issue_slice_load(u, su, s + 2, oy, ox, tid);
            if (s + 3 < NV)
                prefetch_slice(u, s + 3, oy, ox, tid);

            // gradient-magnitude fields over the 18x18 halo region
            const int sl0 = s % 3;
            const int sl1 = (s + 1) % 3;
#pragma unroll
            for (int k = 0; k < 2; ++k) {
                int i = tid + k * 256;
                if (i < 18 * 18) {
                    int r  = i / 18;
                    int cc = i - r * 18;
                    int gy = oy - 1 + r;
                    int gx = ox - 1 + cc;
                    bool gvalid = (gy >= 0) && (gy < NG) && (gx >= 0) && (gx < NG);
#pragma unroll
                    for (int c = 0; c < 3; ++c) {
                        float g0 = 0.0f, g1 = 0.0f, g2 = 0.0f;
                        if (gvalid) {
                            const float* uc0 = su + ((c * 3 + sl0) * 19) * USTR; // slice s
                            const float* uc1 = su + ((c * 3 + sl1) * 19) * USTR; // slice s+1
                            float a = uc0[r * USTR + cc];
                            g0 = fabsf(uc1[r * USTR + cc]       - a);   // D-shift
                            g1 = fabsf(uc0[(r + 1) * USTR + cc] - a);   // H-shift
                            g2 = fabsf(uc0[r * USTR + cc + 1]   - a);   // W-shift
                        }
                        sg[(c * 3 + 0) * 18 * GSTR + r * GSTR + cc] = g0;
                        sg[(c * 3 + 1) * 18 * GSTR + r * GSTR + cc] = g1;
                        sg[(c * 3 + 2) * 18 * GSTR + r * GSTR + cc] = g2;
                    }
                }
            }
            __syncthreads();

            // 3x3 (y,x) sum centered at this thread's output position
#pragma unroll
            for (int f = 0; f < 9; ++f) {
                const float* gf = sg + f * 18 * GSTR;
                float h = 0.0f;
#pragma unroll
                for (int dr = 0; dr < 3; ++dr) {
                    const float* row = gf + (ty + dr) * GSTR + tx;
                    h += row[0] + row[1] + row[2];
                }
                Hc[f] = h;
            }
        } else {
#pragma unroll
            for (int f = 0; f < 9; ++f) Hc[f] = 0.0f;
        }

        // emit output plane z = s-1 using H_{z-1}=Hm2, H_z=Hm1, H_{z+1}=Hc
        const int z = s - 1;
        if (z >= z0 && z <= zEnd && outValid) {
            float S[9];
#pragma unroll
            for (int f = 0; f < 9; ++f)
                S[f] = (Hm2[f] + Hm1[f] + Hc[f]) * (1.0f / 27.0f);

            // f = c*3 + d ; d: 0=D-shift(fdy), 1=H-shift(fdx), 2=W-shift(fdz)
            const float dydy = S[0], dydx = S[1], dydz = S[2];   // channel 0
            const float dxdy = S[3], dxdx = S[4], dxdz = S[5];   // channel 1
            const float dzdy = S[6], dzdx = S[7], dzdz = S[8];   // channel 2

            const float det1 = (dxdx + 1.0f) * ((dydy + 1.0f) * (dzdz + 1.0f) - dydz * dzdy);
            const float det2 = -dxdy * (dydx * (dzdz + 1.0f) - dydz * dzdx);
            const float det3 =  dxdz * (dydx * dzdy - (dydy + 1.0f) * dzdx);
            __builtin_nontemporal_store(det1 + det2 + det3,
                                        &out[((size_t)z * NG + y) * NG + x]);
        }

#pragma unroll
        for (int f = 0; f < 9; ++f) { Hm2[f] = Hm1[f]; Hm1[f] = Hc[f]; }
    }
}

extern "C" void kernel_launch(void* const* d_in, const int* in_sizes, int n_in,
                              void* d_out, int out_size, void* d_ws, size_t ws_size,
                              hipStream_t stream)
{
    (void)in_sizes; (void)n_in; (void)out_size; (void)d_ws; (void)ws_size;
    const float* u   = (const float*)d_in[0];
    float*       out = (float*)d_out;

    dim3 grid((NG + TS - 1) / TS,      // 12
              (NG + TS - 1) / TS,      // 12
              (NG + CH - 1) / CH);     // 8
    dim3 block(TS, TS, 1);             // 256 threads = 8 waves (wave32)
    jacobian_det_kernel<<<grid, block, 0, stream>>>(u, out);
}